// DPLayer_87900800680222
// MI455X (gfx1250) — compile-verified
//
#include <hip/hip_runtime.h>
#include <cstdint>
#include <cstddef>

// Problem constants from the reference
#define BATCH 512
#define HH 256
#define WW 256
#define NULLV 1e30f   // MIN-op DP, NULL = +1e30
#define DEPTH 4       // async prefetch pipeline depth (row buffers)

// Builtin signature (probe-confirmed): (as(1) int* gaddr, as(3) int* ldsaddr, imm, imm)
typedef __attribute__((address_space(1))) int gbl_i32;
typedef __attribute__((address_space(3))) int lds_i32;

#if defined(__has_builtin)
#if __has_builtin(__builtin_amdgcn_global_load_async_to_lds_b32)
#define HAVE_ASYNC_BUILTIN 1
#endif
#if __has_builtin(__builtin_amdgcn_s_wait_asynccnt)
#define HAVE_WAIT_ASYNC_BUILTIN 1
#endif
#endif

// ---- CDNA5 async global -> LDS copy (1 dword per lane) --------------------
// dep: value whose register-readiness orders this issue after the DS read of
// the staging slot being recycled (prevents async write-before-read race).
__device__ __forceinline__ void async_row_b32(const float* gp, float* lp, float dep) {
#ifdef HAVE_ASYNC_BUILTIN
  (void)dep;
  // Ensure the prior ds_load of this staging slot has fully retired.
  asm volatile("s_wait_dscnt 0x0" ::: "memory");
  __builtin_amdgcn_global_load_async_to_lds_b32((gbl_i32*)gp, (lds_i32*)lp, 0, 0);
#else
  // Low 32 bits of a generic LDS pointer are the LDS byte offset on AMDGCN.
  unsigned lofs = (unsigned)(uintptr_t)lp;
  unsigned long long ga = (unsigned long long)(uintptr_t)gp;
  asm volatile("global_load_async_to_lds_b32 %0, %1, off"
               :: "v"(lofs), "v"(ga), "v"(dep)
               : "memory");
#endif
}

// Async loads complete in order: waiting for <= DEPTH-1 outstanding guarantees
// the oldest (current row) has landed in LDS.
__device__ __forceinline__ void wait_async_steady() {
#ifdef HAVE_WAIT_ASYNC_BUILTIN
  __builtin_amdgcn_s_wait_asynccnt(DEPTH - 1);
#else
  asm volatile("s_wait_asynccnt 0x3" ::: "memory");
#endif
}

// ---- wave32 inclusive scans (Hillis-Steele via shuffles) ------------------
__device__ __forceinline__ float wave_scan_add(float v, int lane) {
#pragma unroll
  for (int d = 1; d < 32; d <<= 1) {
    float t = __shfl_up(v, (unsigned)d, 32);
    if (lane >= d) v += t;
  }
  return v;
}
__device__ __forceinline__ float wave_scan_min(float v, int lane) {
#pragma unroll
  for (int d = 1; d < 32; d <<= 1) {
    float t = __shfl_up(v, (unsigned)d, 32);
    if (lane >= d) v = fminf(v, t);
  }
  return v;
}

// V[i][j] = theta[i][j] + min(V[i][j-1], V[i-1][j], V[i-1][j-1]) via
// V = S + prefix_min(min(Vprev, shift(Vprev)) - S_excl), S = cumsum(theta_row).
__global__ __launch_bounds__(WW) void dp_min_path_kernel(const float* __restrict__ x,
                                                         float* __restrict__ out) {
  __shared__ float s_row[DEPTH][WW];    // async-filled row ring buffer
  __shared__ float s_v[WW];             // previous-row V (for the j-1 shift)
  __shared__ float s_agg_add[WW / 32];  // per-wave scan aggregates
  __shared__ float s_agg_min[WW / 32];

  const int j    = threadIdx.x;  // column
  const int wv   = j >> 5;       // wave id within block (8 waves)
  const int lane = j & 31;
  const int b    = blockIdx.x;   // batch

  const float* src = x + (size_t)b * HH * WW + j;

  // Prime the pipeline: rows 0..DEPTH-1 in flight (ASYNCcnt = DEPTH).
#pragma unroll
  for (int r = 0; r < DEPTH; ++r) async_row_b32(src + (size_t)r * WW, &s_row[r][j], 0.0f);

  float V = 0.0f;
  for (int i = 0; i < HH; ++i) {
    wait_async_steady();  // row i has landed once <= DEPTH-1 remain in flight

    const int buf = i & (DEPTH - 1);
    float xv = s_row[buf][j];
    // stable softplus: max(x,0) + log1p(exp(-|x|))
    float th = fmaxf(xv, 0.0f) + log1pf(expf(-fabsf(xv)));

    // Unconditionally recycle this buffer for row i+DEPTH (clamped to the last
    // row; the clamped tail issues target buffers that are never read again,
    // so they are harmless and keep the wait immediate constant).
    int pre = i + DEPTH;
    if (pre > HH - 1) pre = HH - 1;
    async_row_b32(src + (size_t)pre * WW, &s_row[buf][j], th);

    // ---- block-wide inclusive prefix sum of th -> S ----
    float s = wave_scan_add(th, lane);
    if (lane == 31) s_agg_add[wv] = s;
    __syncthreads();  // B1
    float offs = 0.0f;
    for (int w = 0; w < wv; ++w) offs += s_agg_add[w];
    const float S     = s + offs;
    const float Sexcl = S - th;

    if (i == 0) {
      V = S;          // row 0: plain cumsum
      s_v[j] = V;
      __syncthreads();  // close row 0 so row 1's agg writes can't race our reads
      continue;
    }

    // ---- A = min(Vprev, shift(Vprev)); U = A - Sexcl; prefix-min of U ----
    const float vsh = (j == 0) ? NULLV : s_v[j - 1];
    const float U   = fminf(V, vsh) - Sexcl;

    float m = wave_scan_min(U, lane);
    if (lane == 31) s_agg_min[wv] = m;
    __syncthreads();  // B2
    float mo = NULLV;
    for (int w = 0; w < wv; ++w) mo = fminf(mo, s_agg_min[w]);
    const float M = fminf(m, mo);

    V = S + M;
    s_v[j] = V;  // safe: next read of s_v is after B1 of the next row
  }

  if (j == WW - 1) out[b] = V;  // vlast[:, -1]
}

extern "C" void kernel_launch(void* const* d_in, const int* in_sizes, int n_in,
                              void* d_out, int out_size, void* d_ws, size_t ws_size,
                              hipStream_t stream) {
  (void)in_sizes; (void)n_in; (void)d_ws; (void)ws_size; (void)out_size;
  const float* images = (const float*)d_in[0];  // (512, 1, 256, 256) f32
  float* out = (float*)d_out;                   // (512,) f32
  dp_min_path_kernel<<<BATCH, WW, 0, stream>>>(images, out);
}